// RPN_3D_loss_78469052498702
// MI455X (gfx1250) — compile-verified
//
#include <hip/hip_runtime.h>
#include <hip/hip_bf16.h>
#include <math.h>

// ---- problem constants (match reference) ----
#define FEAT_H 32
#define FEAT_W 110
#define STRIDEPX 16
#define NB 8
#define NG 32
#define NANC 36
#define NC 4
#define NANCH (FEAT_H * FEAT_W * NANC)   // 126720
#define FG_T 0.5f
#define BG_LO 0.0f
#define BG_HI 0.5f
#define BEST_T 0.35f

#define BLK 256
#define BLOCKS_PER_B (NANCH / BLK)       // 495
#define NBLOCKS (NB * BLOCKS_PER_B)      // 3960 (divisible by 4 for WMMA chunks)

// partial rows padded to 16 floats so every lane of the WMMA B-operand load
// is unconditional (no EXEC masking around v_wmma)
#define PROW 16

// ws layout: [0,2048) u64 best-keys[NB*NG]; [2048, 2048+NBLOCKS*PROW*4) f32 partials
#define WS_PARTIALS_OFF 2048

typedef __attribute__((ext_vector_type(2))) float v2f;
typedef __attribute__((ext_vector_type(8))) float v8f;

__device__ __forceinline__ float iou_f(float rx1, float ry1, float rx2, float ry2,
                                       float gx1, float gy1, float gx2, float gy2) {
    float x1 = fmaxf(rx1, gx1), y1 = fmaxf(ry1, gy1);
    float x2 = fminf(rx2, gx2), y2 = fminf(ry2, gy2);
    float iw = fmaxf(x2 - x1 + 1.0f, 0.0f);
    float ih = fmaxf(y2 - y1 + 1.0f, 0.0f);
    float inter = iw * ih;
    float ar = (rx2 - rx1 + 1.0f) * (ry2 - ry1 + 1.0f);
    float ag = (gx2 - gx1 + 1.0f) * (gy2 - gy1 + 1.0f);
    return inter / (ar + ag - inter);
}

__device__ __forceinline__ float sl1(float x) {
    float ax = fabsf(x);
    return ax < 1.0f ? 0.5f * ax * ax : ax - 0.5f;
}

// -------- pass 0: zero the per-(b,g) best keys --------
__global__ void k_init(unsigned long long* __restrict__ keys) {
    int i = threadIdx.x;
    if (i < NB * NG) keys[i] = 0ull;
}

// -------- pass 1: per-GT best anchor via packed u64 atomicMax --------
// key = (f32bits(iou) << 32) | (0xFFFFFFFF - anchor)  -> max == highest IoU,
// ties -> smallest anchor index (matches JAX argmax-first). Only anchors with
// iou >= BEST_T participate (force rule needs gt_best >= BEST_T anyway).
__global__ __launch_bounds__(BLK)
void k_best(const float* __restrict__ gtb, const unsigned char* __restrict__ gval,
            const float* __restrict__ anc, unsigned long long* __restrict__ keys) {
    __shared__ float sgtb[NG * 4];
    __shared__ unsigned char sval[NG];
    int tid = threadIdx.x;
    int b = blockIdx.x / BLOCKS_PER_B;
    int chunk = blockIdx.x % BLOCKS_PER_B;
    int a = chunk * BLK + tid;

    if (tid < NG * 4) sgtb[tid] = gtb[b * NG * 4 + tid];
    if (tid < NG) sval[tid] = gval[b * NG + tid];
    __syncthreads();

    int cell = a / NANC;
    int na = a - cell * NANC;
    int cy = cell / FEAT_W;
    int cx = cell - cy * FEAT_W;
    float sx = (float)(cx * STRIDEPX), sy = (float)(cy * STRIDEPX);
    float rx1 = anc[na * 9 + 0] + sx;
    float ry1 = anc[na * 9 + 1] + sy;
    float rx2 = anc[na * 9 + 2] + sx;
    float ry2 = anc[na * 9 + 3] + sy;

    for (int g = 0; g < NG; ++g) {
        if (!sval[g]) continue;
        float ov = iou_f(rx1, ry1, rx2, ry2,
                         sgtb[g * 4 + 0], sgtb[g * 4 + 1], sgtb[g * 4 + 2], sgtb[g * 4 + 3]);
        if (ov >= BEST_T) {
            unsigned long long key =
                ((unsigned long long)__float_as_uint(ov) << 32) |
                (unsigned long long)(0xFFFFFFFFu - (unsigned)a);
            atomicMax(&keys[b * NG + g], key);
        }
    }
}

// -------- pass 2: assignment + per-anchor loss, per-block partials --------
__global__ __launch_bounds__(BLK)
void k_main(const float* __restrict__ cls, const float* __restrict__ b2d,
            const float* __restrict__ b3d, const float* __restrict__ gtb,
            const float* __restrict__ gt3, const int* __restrict__ glbl,
            const unsigned char* __restrict__ gval, const float* __restrict__ anc,
            const float* __restrict__ means, const float* __restrict__ stds,
            const unsigned long long* __restrict__ keys, float* __restrict__ partials) {
    __shared__ float sgtb[NG * 4];
    __shared__ float sgt3[NG * 7];
    __shared__ int slbl[NG];
    __shared__ unsigned char sval[NG];
    __shared__ unsigned long long skey[NG];
    __shared__ float sanc[NANC * 9];
    __shared__ float smean[11];
    __shared__ float sstd[11];
    __shared__ float sred[5 * 8];

    int tid = threadIdx.x;
    int b = blockIdx.x / BLOCKS_PER_B;
    int chunk = blockIdx.x % BLOCKS_PER_B;
    int a = chunk * BLK + tid;
    size_t aidx = (size_t)b * NANCH + (size_t)a;

    // CDNA5 prefetch of this thread's cls row into the cache hierarchy
    __builtin_prefetch(cls + aidx * 4, 0, 0);

    // Stage GT boxes for this batch with the CDNA5 async global->LDS path:
    // 32 lanes x B128 = 512 bytes = gt_boxes[b][:][:]
    if (tid < 32) {
        unsigned long long gaddr = (unsigned long long)(const void*)(gtb + (size_t)b * NG * 4)
                                   + (unsigned long long)tid * 16ull;
        unsigned ldsoff = (unsigned)(unsigned long long)(const void*)(&sgtb[0]) + (unsigned)tid * 16u;
        asm volatile("global_load_async_to_lds_b128 %0, %1, off"
                     :: "v"(ldsoff), "v"(gaddr) : "memory");
        asm volatile("s_wait_asynccnt 0x0" ::: "memory");
    }
    if (tid < NG * 7) sgt3[tid] = gt3[b * NG * 7 + tid];
    if (tid < NG) {
        slbl[tid] = glbl[b * NG + tid];
        sval[tid] = gval[b * NG + tid];
        skey[tid] = keys[b * NG + tid];
    }
    if (tid < NANC * 9) sanc[tid] = anc[tid];
    if (tid >= BLK - 32) {
        int j = tid - (BLK - 32);
        if (j < 11) { smean[j] = means[j]; sstd[j] = stds[j]; }
    }
    __syncthreads();

    // roi geometry
    int cell = a / NANC;
    int na = a - cell * NANC;
    int cy = cell / FEAT_W;
    int cx = cell - cy * FEAT_W;
    float sx = (float)(cx * STRIDEPX), sy = (float)(cy * STRIDEPX);
    float rx1 = sanc[na * 9 + 0] + sx;
    float ry1 = sanc[na * 9 + 1] + sy;
    float rx2 = sanc[na * 9 + 2] + sx;
    float ry2 = sanc[na * 9 + 3] + sy;
    float w = rx2 - rx1 + 1.0f;
    float h = ry2 - ry1 + 1.0f;
    float ctrx = rx1 + 0.5f * w;
    float ctry = ry1 + 0.5f * h;

    // IoU row: masked max / first-argmax
    float best = -INFINITY;
    int agt = 0;
    for (int g = 0; g < NG; ++g) {
        float ov = iou_f(rx1, ry1, rx2, ry2,
                         sgtb[g * 4 + 0], sgtb[g * 4 + 1], sgtb[g * 4 + 2], sgtb[g * 4 + 3]);
        float val = sval[g] ? ov : -1.0f;
        if (val > best) { best = val; agt = g; }
    }
    bool fg = best >= FG_T;
    // best-anchor-per-gt override (key != 0 implies valid && iou >= BEST_T)
    for (int g = 0; g < NG; ++g) {
        unsigned long long k = skey[g];
        if (k != 0ull) {
            unsigned besta = 0xFFFFFFFFu - (unsigned)(k & 0xFFFFFFFFull);
            if (besta == (unsigned)a) { fg = true; agt = g; }
        }
    }
    bool bg = (!fg) && (best < BG_HI) && (best >= BG_LO);
    bool active = fg || bg;
    int lbl = fg ? slbl[agt] : 0;

    float acc_ce = 0.0f, acc2 = 0.0f, acc3 = 0.0f, acc_act = 0.0f, acc_fg = 0.0f;

    if (active) {
        acc_act = 1.0f;
        const float4 c4 = *(const float4*)(cls + aidx * 4);
        float m = fmaxf(fmaxf(c4.x, c4.y), fmaxf(c4.z, c4.w));
        float se = expf(c4.x - m) + expf(c4.y - m) + expf(c4.z - m) + expf(c4.w - m);
        float lse = m + logf(se);
        float xl = (lbl == 0) ? c4.x : (lbl == 1) ? c4.y : (lbl == 2) ? c4.z : c4.w;
        acc_ce = lse - xl;
    }
    if (fg) {
        acc_fg = 1.0f;
        float gx1 = sgtb[agt * 4 + 0], gy1 = sgtb[agt * 4 + 1];
        float gx2 = sgtb[agt * 4 + 2], gy2 = sgtb[agt * 4 + 3];
        float gw = gx2 - gx1 + 1.0f, gh = gy2 - gy1 + 1.0f;
        float gcx = gx1 + 0.5f * gw, gcy = gy1 + 0.5f * gh;
        float t2[4];
        t2[0] = (gcx - ctrx) / w;
        t2[1] = (gcy - ctry) / h;
        t2[2] = logf(gw / w);
        t2[3] = logf(gh / h);
        float t3[7];
        const float* g3 = &sgt3[agt * 7];
        t3[0] = (g3[0] - ctrx) / w;
        t3[1] = (g3[1] - ctry) / h;
        t3[2] = g3[2] - sanc[na * 9 + 4];
        t3[3] = logf(g3[3] / sanc[na * 9 + 5]);
        t3[4] = logf(g3[4] / sanc[na * 9 + 6]);
        t3[5] = logf(g3[5] / sanc[na * 9 + 7]);
        t3[6] = g3[6] - sanc[na * 9 + 8];

        const float4 p2 = *(const float4*)(b2d + aidx * 4);
        float p2a[4] = {p2.x, p2.y, p2.z, p2.w};
#pragma unroll
        for (int j = 0; j < 4; ++j) {
            float t = (t2[j] - smean[j]) / sstd[j];
            acc2 += sl1(p2a[j] - t);
        }
        size_t base3 = aidx * 7;
#pragma unroll
        for (int j = 0; j < 7; ++j) {
            float t = (t3[j] - smean[4 + j]) / sstd[4 + j];
            acc3 += sl1(b3d[base3 + j] - t);
        }
    }

    // wave32 shuffle reduction, then cross-wave via LDS (deterministic)
#pragma unroll
    for (int off = 16; off > 0; off >>= 1) {
        acc_ce  += __shfl_down(acc_ce, off, 32);
        acc2    += __shfl_down(acc2, off, 32);
        acc3    += __shfl_down(acc3, off, 32);
        acc_act += __shfl_down(acc_act, off, 32);
        acc_fg  += __shfl_down(acc_fg, off, 32);
    }
    int wid = tid >> 5, lane = tid & 31;
    if (lane == 0) {
        sred[0 * 8 + wid] = acc_ce;
        sred[1 * 8 + wid] = acc2;
        sred[2 * 8 + wid] = acc3;
        sred[3 * 8 + wid] = acc_act;
        sred[4 * 8 + wid] = acc_fg;
    }
    __syncthreads();
    if (tid == 0) {
        float r[PROW];
#pragma unroll
        for (int j = 0; j < PROW; ++j) r[j] = 0.0f;
#pragma unroll
        for (int wv = 0; wv < 8; ++wv) {
            r[0] += sred[0 * 8 + wv];
            r[1] += sred[1 * 8 + wv];
            r[2] += sred[2 * 8 + wv];
            r[3] += sred[3 * 8 + wv];
            r[4] += sred[4 * 8 + wv];
        }
        float* P = partials + (size_t)blockIdx.x * PROW;
#pragma unroll
        for (int j = 0; j < PROW; ++j) P[j] = r[j];
    }
}

// -------- pass 3: WMMA-chained column-sum of partials, final scalar --------
// S[n] = sum_r P[r][n] computed as D = ones(16x4) x B(4x16) + C accumulation.
// B layout for V_WMMA_F32 (4x16): lanes 0-15 -> N=lane, VGPR0 K=0 / VGPR1 K=1;
// lanes 16-31 -> N=lane-16, VGPR0 K=2 / VGPR1 K=3. D: VGPR0 lane n = D[0][n].
// Rows padded to PROW=16 so every lane loads unconditionally (EXEC stays all-1).
__global__ __launch_bounds__(32)
void k_final(const float* __restrict__ P, float* __restrict__ out) {
    int lane = threadIdx.x;
    int n = lane & 15;
    int hi = lane >> 4;                      // 0 or 1
    v2f aones; aones.x = 1.0f; aones.y = 1.0f;
    v8f c = {};
    for (int base = 0; base < NBLOCKS; base += 4) {
        int r0 = base + 2 * hi;
        float bx = P[(size_t)r0 * PROW + n];
        float by = P[(size_t)r0 * PROW + PROW + n];
        v2f bm; bm.x = bx; bm.y = by;
        c = __builtin_amdgcn_wmma_f32_16x16x4_f32(
            /*neg_a=*/false, aones, /*neg_b=*/false, bm,
            /*c_mod=*/(short)0, c, /*reuse_a=*/false, /*reuse_b=*/false);
    }
    float c0 = c[0];
    float sce  = __shfl(c0, 0, 32);
    float s2   = __shfl(c0, 1, 32);
    float s3   = __shfl(c0, 2, 32);
    float nact = __shfl(c0, 3, 32);
    float nfg  = __shfl(c0, 4, 32);
    if (lane == 0) {
        out[0] = sce / fmaxf(nact, 1.0f)
               + s2 / fmaxf(nfg, 1.0f)
               + s3 / fmaxf(nfg, 1.0f);
    }
}

extern "C" void kernel_launch(void* const* d_in, const int* in_sizes, int n_in,
                              void* d_out, int out_size, void* d_ws, size_t ws_size,
                              hipStream_t stream) {
    const float* cls  = (const float*)d_in[0];
    const float* b2d  = (const float*)d_in[1];
    const float* b3d  = (const float*)d_in[2];
    const float* gtb  = (const float*)d_in[3];
    const float* gt3  = (const float*)d_in[4];
    const int* glbl   = (const int*)d_in[5];
    const unsigned char* gval = (const unsigned char*)d_in[6];
    const float* anc  = (const float*)d_in[7];
    const float* means = (const float*)d_in[8];
    const float* stds  = (const float*)d_in[9];

    unsigned long long* keys = (unsigned long long*)d_ws;
    float* partials = (float*)((char*)d_ws + WS_PARTIALS_OFF);
    float* out = (float*)d_out;

    hipLaunchKernelGGL(k_init, dim3(1), dim3(256), 0, stream, keys);
    hipLaunchKernelGGL(k_best, dim3(NBLOCKS), dim3(BLK), 0, stream, gtb, gval, anc, keys);
    hipLaunchKernelGGL(k_main, dim3(NBLOCKS), dim3(BLK), 0, stream,
                       cls, b2d, b3d, gtb, gt3, glbl, gval, anc, means, stds,
                       keys, partials);
    hipLaunchKernelGGL(k_final, dim3(1), dim3(32), 0, stream, partials, out);
}